// ChannelAttentionModule_85487029060053
// MI455X (gfx1250) — compile-verified
//
#include <hip/hip_runtime.h>
#include <hip/hip_bf16.h>
#include <math.h>

// Problem constants (from reference)
#define GROUPS 11
#define CHANNELS 704
#define GW 64
#define HID 16
#define BATCH 128
#define LEN 1024
#define NROWS (BATCH * CHANNELS)   // 90112
#define MSTACK (2 * BATCH)         // 256 stacked rows (avg | max)

typedef __attribute__((ext_vector_type(2))) float v2f;
typedef __attribute__((ext_vector_type(8))) float v8f;

// ---------------------------------------------------------------------------
// Kernel 1: per-(b,c) row reduction over L=1024 -> mean and max.
// One wave32 per row; 8 coalesced float4 loads per lane; shfl_xor tree reduce.
// Output: ystack[256][704] f32, row b = mean, row 128+b = max.
// ---------------------------------------------------------------------------
__global__ __launch_bounds__(256) void ca_reduce_kernel(
    const float* __restrict__ x, float* __restrict__ ystack) {
  const int wave = threadIdx.x >> 5;
  const int lane = threadIdx.x & 31;
  const int row  = blockIdx.x * 8 + wave;        // 11264 * 8 == 90112 exactly

  const float4* __restrict__ xr =
      (const float4*)(x + (size_t)row * LEN);

  float s = 0.0f;
  float m = -INFINITY;
#pragma unroll
  for (int i = 0; i < 8; ++i) {
    float4 v = xr[lane + i * 32];                // 512B contiguous per wave-iter
    s += v.x + v.y + v.z + v.w;
    m = fmaxf(m, fmaxf(fmaxf(v.x, v.y), fmaxf(v.z, v.w)));
  }
#pragma unroll
  for (int off = 16; off > 0; off >>= 1) {
    s += __shfl_xor(s, off, 32);
    m = fmaxf(m, __shfl_xor(m, off, 32));
  }
  if (lane == 0) {
    const int b = row / CHANNELS;
    const int c = row % CHANNELS;
    ystack[(size_t)b * CHANNELS + c]           = s * (1.0f / (float)LEN);
    ystack[(size_t)(BATCH + b) * CHANNELS + c] = m;
  }
}

// ---------------------------------------------------------------------------
// Kernel 2: grouped bottleneck MLP via V_WMMA_F32_16X16X4_F32 (full f32).
//   GEMM1: A[16x64] (ystack tile, group g cols) x W1[g][64x16] -> h[16x16]
//   GEMM2: h[16x16] x W2[g][16x64] -> o[16x64]
// One wave per (group, M-tile). 16 + 16 = 32 WMMAs per wave.
// A-frag layout (ISA 7.12.2, 32-bit A 16x4): lanes 0-15 hold K={0,1} for M=lane,
// lanes 16-31 hold K={2,3} for M=lane-16. B 4x16 mirrored. C/D: VGPR r holds
// (M=r, N=lane) for lanes 0-15 and (M=r+8, N=lane-16) for lanes 16-31.
// ---------------------------------------------------------------------------
__global__ __launch_bounds__(32) void ca_mlp_wmma_kernel(
    const float* __restrict__ ystack,
    const float* __restrict__ W1,     // [G][GW][HID] row-major
    const float* __restrict__ W2,     // [G][HID][GW] row-major
    float* __restrict__ omlp) {       // [256][704]
  const int g    = blockIdx.x;        // 0..10
  const int t    = blockIdx.y;        // 0..15  (M-tile of the 256 stacked rows)
  const int lane = threadIdx.x;       // 0..31
  const int half = lane >> 4;         // 0: K pair {0,1}; 1: K pair {2,3}
  const int lid  = lane & 15;         // row/col index within half

  __shared__ float hbuf[16][17];      // 16x16 hidden tile, padded vs bank conflicts

  const int m0 = t * 16;
  const float* __restrict__ Ag  = ystack + (size_t)m0 * CHANNELS + g * GW;
  const float* __restrict__ W1g = W1 + (size_t)g * (GW * HID);
  const float* __restrict__ W2g = W2 + (size_t)g * (HID * GW);

  // ---- GEMM1: [16x64] x [64x16] as 16 steps of K=4 ----
  v8f acc = {0.f, 0.f, 0.f, 0.f, 0.f, 0.f, 0.f, 0.f};
#pragma unroll
  for (int kk = 0; kk < 16; ++kk) {
    const int k0 = kk * 4 + half * 2;
    v2f a, b;
    a.x = Ag[(size_t)lid * CHANNELS + k0];
    a.y = Ag[(size_t)lid * CHANNELS + k0 + 1];
    b.x = W1g[(k0    ) * HID + lid];
    b.y = W1g[(k0 + 1) * HID + lid];
    acc = __builtin_amdgcn_wmma_f32_16x16x4_f32(
        /*neg_a=*/false, a, /*neg_b=*/false, b,
        /*c_mod=*/(short)0, acc, /*reuse_a=*/false, /*reuse_b=*/false);
  }

  // Spill hidden tile (C/D layout) to LDS so it can be re-read in A layout.
#pragma unroll
  for (int r = 0; r < 8; ++r)
    hbuf[r + half * 8][lid] = acc[r];
  __syncthreads();

  // ---- GEMM2: [16x16] x [16x64] -> four 16x16 output tiles, 4 K-steps each ----
#pragma unroll
  for (int nt = 0; nt < 4; ++nt) {
    v8f acc2 = {0.f, 0.f, 0.f, 0.f, 0.f, 0.f, 0.f, 0.f};
#pragma unroll
    for (int kk = 0; kk < 4; ++kk) {
      const int k0 = kk * 4 + half * 2;
      v2f a, b;
      a.x = hbuf[lid][k0];
      a.y = hbuf[lid][k0 + 1];
      b.x = W2g[(k0    ) * GW + nt * 16 + lid];
      b.y = W2g[(k0 + 1) * GW + nt * 16 + lid];
      acc2 = __builtin_amdgcn_wmma_f32_16x16x4_f32(
          false, a, false, b, (short)0, acc2, false, false);
    }
#pragma unroll
    for (int r = 0; r < 8; ++r)
      omlp[(size_t)(m0 + r + half * 8) * CHANNELS + g * GW + nt * 16 + lid] =
          acc2[r];
  }
}

// ---------------------------------------------------------------------------
// Kernel 3: gate = sigmoid(o_avg + o_max); out = x * gate. One block per row,
// 256 threads x float4 = 1024 elements. This pass carries the bandwidth bill.
// ---------------------------------------------------------------------------
__global__ __launch_bounds__(256) void ca_scale_kernel(
    const float* __restrict__ x, const float* __restrict__ omlp,
    float* __restrict__ out) {
  const int row = blockIdx.x;                 // 0..90111 == (b, c)
  const int b = row / CHANNELS;
  const int c = row % CHANNELS;

  const float z = omlp[(size_t)b * CHANNELS + c] +
                  omlp[(size_t)(BATCH + b) * CHANNELS + c];
  const float gate = 1.0f / (1.0f + __expf(-z));

  const float4* __restrict__ xr = (const float4*)(x + (size_t)row * LEN);
  float4* __restrict__ orow     = (float4*)(out + (size_t)row * LEN);

  float4 v = xr[threadIdx.x];
  v.x *= gate; v.y *= gate; v.z *= gate; v.w *= gate;
  orow[threadIdx.x] = v;
}

// ---------------------------------------------------------------------------
extern "C" void kernel_launch(void* const* d_in, const int* in_sizes, int n_in,
                              void* d_out, int out_size, void* d_ws, size_t ws_size,
                              hipStream_t stream) {
  (void)in_sizes; (void)n_in; (void)out_size; (void)ws_size;
  const float* x  = (const float*)d_in[0];   // [128,704,1024]
  const float* W1 = (const float*)d_in[1];   // [11,64,16]
  const float* W2 = (const float*)d_in[2];   // [11,16,64]
  float* out = (float*)d_out;

  float* ystack = (float*)d_ws;                          // [256][704]
  float* omlp   = ystack + (size_t)MSTACK * CHANNELS;    // [256][704]

  ca_reduce_kernel<<<NROWS / 8, 256, 0, stream>>>(x, ystack);
  ca_mlp_wmma_kernel<<<dim3(GROUPS, MSTACK / 16), 32, 0, stream>>>(
      ystack, W1, W2, omlp);
  ca_scale_kernel<<<NROWS, 256, 0, stream>>>(x, omlp, out);
}